// GaussianAvatars_8349416424119
// MI455X (gfx1250) — compile-verified
//
#include <hip/hip_runtime.h>

// ---------------------------------------------------------------------------
// GaussianAvatars binding kernel for MI455X (gfx1250).
// Phase 1: per-face frame/quat/scale/center precompute -> d_ws (5 x float4/face).
// Phase 2: bandwidth-bound per-gaussian stream. gfx1250 async global<->LDS
//          copies (ASYNCcnt) stage the 12B-strided arrays as coalesced b128;
//          binding/rotation/face-record gathers issue in parallel with the
//          async copies so the wave waits exactly once before compute.
// ---------------------------------------------------------------------------

#define TILE 256

#if defined(__HIP_DEVICE_COMPILE__) && defined(__gfx1250__) &&                 \
    __has_builtin(__builtin_amdgcn_global_load_async_to_lds_b128) &&           \
    __has_builtin(__builtin_amdgcn_global_store_async_from_lds_b128) &&        \
    __has_builtin(__builtin_amdgcn_s_wait_asynccnt)
#define HAVE_ASYNC 1
#else
#define HAVE_ASYNC 0
#endif

#if HAVE_ASYNC
// The b128 async builtins take typed int4 pointers (addrspace(1) global,
// addrspace(3) LDS) — confirmed by hipcc diagnostics.
typedef int v4i_t __attribute__((ext_vector_type(4)));
typedef __attribute__((address_space(1))) v4i_t* gp128_t;
typedef __attribute__((address_space(3))) v4i_t* lp128_t;
// Generic->AS1: same 64-bit VA. Generic->AS3: low 32 bits of a flat shared
// address are the LDS byte address (aperture bits live in [63:32]).
__device__ __forceinline__ gp128_t as_global(const void* p) {
  return (gp128_t)(unsigned long long)p;
}
__device__ __forceinline__ lp128_t as_lds(const void* p) {
  return (lp128_t)(unsigned)(unsigned long long)p;
}
#endif

// ---------------------------------------------------------------------------
// Phase 1: face precompute. Record layout (5 x float4 = 80B per face):
//   r0 = (qw, qx, qy, qz)          normalized face quaternion
//   r1 = (a0.x, a0.y, a0.z, s)     frame column 0, scale
//   r2 = (a1.x, a1.y, a1.z, log s) frame column 1, log-scale
//   r3 = (a2.x, a2.y, a2.z, 0)     frame column 2
//   r4 = (cx, cy, cz, 0)           face center
// ---------------------------------------------------------------------------
__global__ void __launch_bounds__(256)
face_kernel(const float* __restrict__ verts, const int* __restrict__ faces,
            float4* __restrict__ frec, int nfaces) {
  const int f = blockIdx.x * blockDim.x + threadIdx.x;
  if (f >= nfaces) return;

  const int i0 = faces[f * 3 + 0];
  const int i1 = faces[f * 3 + 1];
  const int i2 = faces[f * 3 + 2];

  const float v0x = verts[i0 * 3 + 0], v0y = verts[i0 * 3 + 1], v0z = verts[i0 * 3 + 2];
  const float v1x = verts[i1 * 3 + 0], v1y = verts[i1 * 3 + 1], v1z = verts[i1 * 3 + 2];
  const float v2x = verts[i2 * 3 + 0], v2y = verts[i2 * 3 + 1], v2z = verts[i2 * 3 + 2];

  const float e1x = v1x - v0x, e1y = v1y - v0y, e1z = v1z - v0z;
  const float e2x = v2x - v0x, e2y = v2y - v0y, e2z = v2z - v0z;

  // a0 = safe_normalize(e1);  s0 = length(e1)  (same clipped sqrt)
  const float l0 = sqrtf(fmaxf(e1x * e1x + e1y * e1y + e1z * e1z, 1e-20f));
  const float a0x = e1x / l0, a0y = e1y / l0, a0z = e1z / l0;

  // a1 = safe_normalize(cross(a0, e2))
  float cx0 = a0y * e2z - a0z * e2y;
  float cy0 = a0z * e2x - a0x * e2z;
  float cz0 = a0x * e2y - a0y * e2x;
  const float l1 = sqrtf(fmaxf(cx0 * cx0 + cy0 * cy0 + cz0 * cz0, 1e-20f));
  const float a1x = cx0 / l1, a1y = cy0 / l1, a1z = cz0 / l1;

  // a2 = -safe_normalize(cross(a1, a0))
  float cx1 = a1y * a0z - a1z * a0y;
  float cy1 = a1z * a0x - a1x * a0z;
  float cz1 = a1x * a0y - a1y * a0x;
  const float l2 = sqrtf(fmaxf(cx1 * cx1 + cy1 * cy1 + cz1 * cz1, 1e-20f));
  const float a2x = -cx1 / l2, a2y = -cy1 / l2, a2z = -cz1 / l2;

  const float s1 = fabsf(e2x * a2x + e2y * a2y + e2z * a2z);
  const float scale = 0.5f * (l0 + s1);
  const float ctx = (v0x + v1x + v2x) * (1.0f / 3.0f);
  const float cty = (v0y + v1y + v2y) * (1.0f / 3.0f);
  const float ctz = (v0z + v1z + v2z) * (1.0f / 3.0f);

  // orient has columns a0,a1,a2 -> m[i][j]: m00=a0.x m01=a1.x m02=a2.x, etc.
  const float m00 = a0x, m01 = a1x, m02 = a2x;
  const float m10 = a0y, m11 = a1y, m12 = a2y;
  const float m20 = a0z, m21 = a1z, m22 = a2z;

  const float q0 = sqrtf(fmaxf(1.0f + m00 + m11 + m22, 0.0f));
  const float q1 = sqrtf(fmaxf(1.0f + m00 - m11 - m22, 0.0f));
  const float q2 = sqrtf(fmaxf(1.0f - m00 + m11 - m22, 0.0f));
  const float q3 = sqrtf(fmaxf(1.0f - m00 - m11 + m22, 0.0f));

  // jnp.argmax: first occurrence of the max (strict > preserves that).
  int idx = 0;
  float best = q0;
  if (q1 > best) { best = q1; idx = 1; }
  if (q2 > best) { best = q2; idx = 2; }
  if (q3 > best) { best = q3; idx = 3; }

  float rw, rx, ry, rz;
  if (idx == 0)      { rw = q0 * q0;   rx = m21 - m12; ry = m02 - m20; rz = m10 - m01; }
  else if (idx == 1) { rw = m21 - m12; rx = q1 * q1;   ry = m10 + m01; rz = m02 + m20; }
  else if (idx == 2) { rw = m02 - m20; rx = m10 + m01; ry = q2 * q2;   rz = m12 + m21; }
  else               { rw = m10 - m01; rx = m20 + m02; ry = m21 + m12; rz = q3 * q3;   }

  const float denom = 2.0f * fmaxf(best, 0.1f);
  float qw = rw / denom, qx = rx / denom, qy = ry / denom, qz = rz / denom;

  // Pre-normalize (reference normalizes the gathered copy; values identical).
  const float qn = sqrtf(qw * qw + qx * qx + qy * qy + qz * qz);
  const float inv = 1.0f / fmaxf(qn, 1e-12f);
  qw *= inv; qx *= inv; qy *= inv; qz *= inv;

  float4* r = frec + (size_t)f * 5;
  r[0] = make_float4(qw, qx, qy, qz);
  r[1] = make_float4(a0x, a0y, a0z, scale);
  r[2] = make_float4(a1x, a1y, a1z, logf(scale));
  r[3] = make_float4(a2x, a2y, a2z, 0.0f);
  r[4] = make_float4(ctx, cty, ctz, 0.0f);
}

// ---------------------------------------------------------------------------
// Phase 2: per-gaussian stream. TILE=256 gaussians per block.
// ---------------------------------------------------------------------------
__global__ void __launch_bounds__(256)
gauss_kernel(const float* __restrict__ xyz, const float* __restrict__ scal,
             const float* __restrict__ rot, const int* __restrict__ binding,
             const float4* __restrict__ frec, float* __restrict__ out, int n) {
  __shared__ float s_xyz[TILE * 3];
  __shared__ float s_scl[TILE * 3];
  __shared__ float s_out[TILE * 10];

  const int t = threadIdx.x;
  const int base = blockIdx.x * TILE;
  const int nv = min(TILE, n - base);
  const int g = base + t;

  // ---- 1) kick off async staging of the 12B-strided arrays ---------------
  {
    const char* gx = (const char*)(xyz + (size_t)base * 3);
    const char* gs = (const char*)(scal + (size_t)base * 3);
    const int bx = nv * 12;  // bytes, multiple of 16 for nv in {256,128}
    const int off = t * 16;
#if HAVE_ASYNC
    if (off < bx)
      __builtin_amdgcn_global_load_async_to_lds_b128(
          as_global(gx + off), as_lds((const char*)s_xyz + off), 0, 0);
    if (off < bx)
      __builtin_amdgcn_global_load_async_to_lds_b128(
          as_global(gs + off), as_lds((const char*)s_scl + off), 0, 0);
#else
    if (off < bx) *(float4*)((char*)s_xyz + off) = *(const float4*)(gx + off);
    if (off < bx) *(float4*)((char*)s_scl + off) = *(const float4*)(gs + off);
#endif
  }

  // ---- 2) overlapping direct loads: binding, rotation, face record -------
  // These are ordinary VMEM loads issued while the async copies are in
  // flight; the 80B face-record gather hits L2 (table is ~0.8MB << 192MB).
  float4 Q = make_float4(0.f, 0.f, 0.f, 0.f);
  float4 A0 = Q, A1 = Q, A2 = Q, C = Q, RR = Q;
  if (t < nv) {
    const int f = binding[g];
    RR = *(const float4*)(rot + (size_t)g * 4);  // aligned b128
    const float4* r = frec + (size_t)f * 5;
    Q = r[0]; A0 = r[1]; A1 = r[2]; A2 = r[3]; C = r[4];
  }

  // ---- 3) single wait point, then block-wide visibility -------------------
#if HAVE_ASYNC
  __builtin_amdgcn_s_wait_asynccnt(0);
#endif
  __syncthreads();

  // ---- 4) compute ---------------------------------------------------------
  if (t < nv) {
    const float x = s_xyz[t * 3 + 0];
    const float y = s_xyz[t * 3 + 1];
    const float z = s_xyz[t * 3 + 2];

    // g_xyz = (a0*x + a1*y + a2*z) * s + c   (columns a0,a1,a2)
    const float s  = A0.w;
    const float ls = A1.w;
    float gx = A0.x * x + A1.x * y + A2.x * z;
    float gy = A0.y * x + A1.y * y + A2.y * z;
    float gz = A0.z * x + A1.z * y + A2.z * z;
    gx = gx * s + C.x;
    gy = gy * s + C.y;
    gz = gz * s + C.z;

    // quaternion multiply: (face quat) * (raw rotation), then sign-fix
    const float aw = Q.x, ax = Q.y, ay = Q.z, az = Q.w;
    const float bw = RR.x, bx = RR.y, by = RR.z, bz = RR.w;
    float ow = aw * bw - ax * bx - ay * by - az * bz;
    float ox = aw * bx + ax * bw + ay * bz - az * by;
    float oy = aw * by - ax * bz + ay * bw + az * bx;
    float oz = aw * bz + ax * by - ay * bx + az * bw;
    if (ow < 0.0f) { ow = -ow; ox = -ox; oy = -oy; oz = -oz; }

    const float s0 = s_scl[t * 3 + 0] + ls;
    const float s1 = s_scl[t * 3 + 1] + ls;
    const float s2 = s_scl[t * 3 + 2] + ls;

    // stride-10 LDS write: gcd(10,64)=2 -> 32 distinct banks, conflict-free
    float* o = s_out + t * 10;
    o[0] = gx; o[1] = gy; o[2] = gz;
    o[3] = ow; o[4] = ox; o[5] = oy; o[6] = oz;
    o[7] = s0; o[8] = s1; o[9] = s2;
  }
  __syncthreads();

  // ---- 5) stream output tile (nv*40 bytes, 16B-aligned) -------------------
  {
    char* gd = (char*)out + (size_t)base * 40;
    const char* lsrc = (const char*)s_out;
    const int ob = nv * 40;
    for (int off = t * 16; off < ob; off += TILE * 16) {
#if HAVE_ASYNC
      __builtin_amdgcn_global_store_async_from_lds_b128(
          as_global(gd + off), as_lds(lsrc + off), 0, 0);
#else
      *(float4*)(gd + off) = *(const float4*)(lsrc + off);
#endif
    }
    // S_ENDPGM performs an implicit wait-idle, draining ASYNCcnt.
  }
}

// ---------------------------------------------------------------------------
extern "C" void kernel_launch(void* const* d_in, const int* in_sizes, int n_in,
                              void* d_out, int out_size, void* d_ws, size_t ws_size,
                              hipStream_t stream) {
  (void)n_in; (void)out_size; (void)ws_size;
  const float* verts   = (const float*)d_in[0];
  const float* xyz     = (const float*)d_in[1];
  const float* scaling = (const float*)d_in[2];
  const float* rot     = (const float*)d_in[3];
  const int*   faces   = (const int*)d_in[4];
  const int*   binding = (const int*)d_in[5];
  float*       out     = (float*)d_out;

  const int nfaces = in_sizes[4] / 3;  // 9976
  const int ngauss = in_sizes[5];      // 2,000,000
  float4* frec = (float4*)d_ws;        // needs nfaces*5*16 B ~= 798 KB

  face_kernel<<<(nfaces + 255) / 256, 256, 0, stream>>>(verts, faces, frec, nfaces);
  gauss_kernel<<<(ngauss + TILE - 1) / TILE, TILE, 0, stream>>>(
      xyz, scaling, rot, binding, frec, out, ngauss);
}